// GCNPredictor_56135222559004
// MI455X (gfx1250) — compile-verified
//
#include <hip/hip_runtime.h>
#include <hip/hip_bf16.h>

typedef __attribute__((ext_vector_type(2))) float v2f;
typedef __attribute__((ext_vector_type(8))) float v8f;

#define HD 128          // hidden/input dim
#define LDA 132         // padded LDS row stride (floats) to avoid bank conflicts

// ---------------- degree / normalization ----------------

__global__ void k_init_deg(float* deg, int n) {
    int i = blockIdx.x * blockDim.x + threadIdx.x;
    if (i < n) deg[i] = 1.0f;   // self-loop contributes weight 1
}

__global__ void k_deg(const int* dst, const float* ew, float* deg, int E) {
    int e = blockIdx.x * blockDim.x + threadIdx.x;
    if (e < E) unsafeAtomicAdd(&deg[dst[e]], ew[e]);
}

__global__ void k_dinv(const float* deg, float* dinv, float* selfc, int n) {
    int i = blockIdx.x * blockDim.x + threadIdx.x;
    if (i < n) {
        float d = fmaxf(deg[i], 1e-12f);
        float r = rsqrtf(d);
        dinv[i]  = r;
        selfc[i] = r * r;
    }
}

__global__ void k_norm(const int* src, const int* dst, const float* ew,
                       const float* dinv, float* norm, int E) {
    int e = blockIdx.x * blockDim.x + threadIdx.x;
    if (e < E) norm[e] = dinv[src[e]] * ew[e] * dinv[dst[e]];
}

// ---------------- fused fp32 WMMA GEMM ----------------
// Y = (relu_in ? relu(X) : X) @ W, then fused epilogue:
//   mode 0 (GCN layer):  XW = Y ; AGG = selfc[row]*Y + bias[col]
//   mode 1 (dense):      OUT = relu(Y + bias[col])        (XW unused)
// One block: 16-row tile staged in LDS (relu applied on load). 8 waves,
// wave w owns cols [16w,16w+16). 32 x V_WMMA_F32_16X16X4_F32 over K=128.
// Requires n % 16 == 0 (50000 = 3125*16).

__global__ __launch_bounds__(256) void k_gemm_fused(
        const float* __restrict__ X, const float* __restrict__ W,
        const float* __restrict__ bias, const float* __restrict__ selfc,
        float* __restrict__ XW, float* __restrict__ AGG,
        int relu_in, int mode, int n) {
    __shared__ float As[16 * LDA];
    const int tid  = threadIdx.x;
    const int row0 = blockIdx.x * 16;

    // cooperative load of the 16x128 A tile, ReLU fused on load
    for (int i = tid; i < 16 * HD; i += 256) {
        int r = i >> 7, c = i & (HD - 1);
        float v = X[(size_t)(row0 + r) * HD + c];
        As[r * LDA + c] = relu_in ? fmaxf(v, 0.f) : v;
    }
    __syncthreads();

    const int lane  = tid & 31;
    const int wv    = tid >> 5;
    const int col0  = wv * 16;
    const int mrow  = lane & 15;             // A row / B col within tile
    const int khalf = (lane >> 4) << 1;      // lanes 16-31 hold K+2,K+3
    const int col   = col0 + mrow;

    v8f acc = {};
#pragma unroll
    for (int k0 = 0; k0 < HD; k0 += 4) {
        const int kk = k0 + khalf;
        v2f a, b;
        // A 16x4 f32 layout: VGPR0 = K=kk, VGPR1 = K=kk+1 (ISA 7.12.2)
        a.x = As[mrow * LDA + kk];
        a.y = As[mrow * LDA + kk + 1];
        // B 4x16 f32: row (K) striped the same way across lane halves
        b.x = W[(size_t)kk * HD + col];
        b.y = W[(size_t)(kk + 1) * HD + col];
        acc = __builtin_amdgcn_wmma_f32_16x16x4_f32(
            /*neg_a=*/false, a, /*neg_b=*/false, b,
            /*c_mod=*/(short)0, acc, /*reuse_a=*/false, /*reuse_b=*/false);
    }

    // C/D layout: VGPR r -> M = r + 8*(lane>=16), N = col0 + (lane&15)
    const int ncol = col0 + (lane & 15);
    const int mb   = row0 + ((lane >> 4) << 3);
    const float bc = bias[ncol];

    if (mode == 0) {
#pragma unroll
        for (int r = 0; r < 8; ++r) {
            const size_t o = (size_t)(mb + r) * HD + ncol;
            const float y = acc[r];
            XW[o]  = y;
            AGG[o] = selfc[mb + r] * y + bc;     // self-loop + bias init
        }
    } else {
#pragma unroll
        for (int r = 0; r < 8; ++r) {
            const size_t o = (size_t)(mb + r) * HD + ncol;
            AGG[o] = fmaxf(acc[r] + bc, 0.f);    // dense: relu(XW + b)
        }
    }
}

// ---------------- edge scatter: AGG[dst] += norm * XW[src] ----------------
// One wave (32 lanes) per edge; each lane handles 4 floats (fp32 atomics
// land in the 192MB L2 since both operand matrices fit).

__global__ void k_scatter(const int* __restrict__ src, const int* __restrict__ dst,
                          const float* __restrict__ norm,
                          const float* __restrict__ xw,
                          float* __restrict__ agg, int E) {
    int t = blockIdx.x * blockDim.x + threadIdx.x;
    int e = t >> 5, lane = t & 31;
    if (e >= E) return;
    int s = src[e], d = dst[e];
    float nm = norm[e];
    float4 v = ((const float4*)(xw + (size_t)s * HD))[lane];
    float* a = agg + (size_t)d * HD + lane * 4;
    unsafeAtomicAdd(a + 0, nm * v.x);
    unsafeAtomicAdd(a + 1, nm * v.y);
    unsafeAtomicAdd(a + 2, nm * v.z);
    unsafeAtomicAdd(a + 3, nm * v.w);
}

// ---------------- heads: out[t] = H[ace[t]].Wace + bace ; out[T+t] likewise --

__global__ void k_heads(const float* __restrict__ Hf,
                        const int* __restrict__ ace, const int* __restrict__ h2,
                        const float* __restrict__ Wace, const float* __restrict__ bace,
                        const float* __restrict__ Wh2, const float* __restrict__ bh2,
                        float* __restrict__ out, int T) {
    int t = blockIdx.x;
    int lane = threadIdx.x & 31;
    int wv = threadIdx.x >> 5;           // wave 0: ace head, wave 1: h2 head
    const int* idx = wv ? h2 : ace;
    const float* Wv = wv ? Wh2 : Wace;
    float bias = wv ? bh2[0] : bace[0];
    int node = idx[t];
    float4 v = ((const float4*)(Hf + (size_t)node * HD))[lane];
    float4 w = ((const float4*)Wv)[lane];
    float s = v.x * w.x + v.y * w.y + v.z * w.z + v.w * w.w;
#pragma unroll
    for (int off = 16; off; off >>= 1) s += __shfl_down(s, off, 32);
    if (lane == 0) out[wv * T + t] = s + bias;
}

// ---------------- launch ----------------

extern "C" void kernel_launch(void* const* d_in, const int* in_sizes, int n_in,
                              void* d_out, int out_size, void* d_ws, size_t ws_size,
                              hipStream_t stream) {
    const float* x    = (const float*)d_in[0];
    const int*   ei   = (const int*)d_in[1];
    const float* ew   = (const float*)d_in[2];
    const int*   ace  = (const int*)d_in[3];
    const int*   h2i  = (const int*)d_in[4];
    const float* W1   = (const float*)d_in[5];  const float* b1 = (const float*)d_in[6];
    const float* W2   = (const float*)d_in[7];  const float* b2 = (const float*)d_in[8];
    const float* W3   = (const float*)d_in[9];  const float* b3 = (const float*)d_in[10];
    const float* Wh   = (const float*)d_in[11]; const float* bh = (const float*)d_in[12];
    const float* Wace = (const float*)d_in[13]; const float* bace = (const float*)d_in[14];
    const float* Wh2  = (const float*)d_in[15]; const float* bh2  = (const float*)d_in[16];
    float* out = (float*)d_out;

    const int n = in_sizes[0] / HD;      // 50000
    const int E = in_sizes[2];           // 640000
    const int T = in_sizes[3];           // 100
    const int* src = ei;
    const int* dst = ei + E;

    // workspace layout (A0/A1 ping-pong aggregation buffers)
    char* ws = (char*)d_ws;
    const size_t NH = (size_t)n * HD * sizeof(float);
    float* A0    = (float*)ws;               ws += NH;
    float* A1    = (float*)ws;               ws += NH;
    float* XW    = (float*)ws;               ws += NH;
    float* deg   = (float*)ws;               ws += (size_t)n * sizeof(float);
    float* dinv  = (float*)ws;               ws += (size_t)n * sizeof(float);
    float* selfc = (float*)ws;               ws += (size_t)n * sizeof(float);
    float* norm  = (float*)ws;               ws += (size_t)E * sizeof(float);

    const int TB = 256;
    const int gN  = (n + TB - 1) / TB;
    const int gE  = (E + TB - 1) / TB;
    const int gSc = (E * 32 + TB - 1) / TB;         // wave per edge
    const int gMM = n / 16;                         // GEMM row tiles

    // normalization (shared across all layers)
    k_init_deg<<<gN, TB, 0, stream>>>(deg, n);
    k_deg<<<gE, TB, 0, stream>>>(dst, ew, deg, E);
    k_dinv<<<gN, TB, 0, stream>>>(deg, dinv, selfc, n);
    k_norm<<<gE, TB, 0, stream>>>(src, dst, ew, dinv, norm, E);

    // layer 1: A0 = selfc*(x@W1)+b1, then scatter; relu deferred to next load
    k_gemm_fused<<<gMM, TB, 0, stream>>>(x,  W1, b1, selfc, XW, A0, 0, 0, n);
    k_scatter<<<gSc, TB, 0, stream>>>(src, dst, norm, XW, A0, E);
    // layer 2: reads relu(A0) on load
    k_gemm_fused<<<gMM, TB, 0, stream>>>(A0, W2, b2, selfc, XW, A1, 1, 0, n);
    k_scatter<<<gSc, TB, 0, stream>>>(src, dst, norm, XW, A1, E);
    // layer 3
    k_gemm_fused<<<gMM, TB, 0, stream>>>(A1, W3, b3, selfc, XW, A0, 1, 0, n);
    k_scatter<<<gSc, TB, 0, stream>>>(src, dst, norm, XW, A0, E);
    // dense trunk: A1 = relu(relu(A0) @ Wh + bh)
    k_gemm_fused<<<gMM, TB, 0, stream>>>(A0, Wh, bh, nullptr, nullptr, A1, 1, 1, n);
    // prediction heads
    k_heads<<<T, 64, 0, stream>>>(A1, ace, h2i, Wace, bace, Wh2, bh2, out, T);
}